// AttentionBlock_17918603559038
// MI455X (gfx1250) — compile-verified
//
#include <hip/hip_runtime.h>
#include <math.h>

typedef __attribute__((ext_vector_type(2))) float v2f;
typedef __attribute__((ext_vector_type(8))) float v8f;
typedef int v4i32 __attribute__((vector_size(16)));   // matches builtin param type

constexpr int kB   = 4;     // batch
constexpr int kC   = 256;   // channels
constexpr int kT   = 4096;  // spatial (64*64)
constexpr int kNH  = 4;     // heads per batch
constexpr int kBH  = 16;    // kB*kNH
constexpr int kCH  = 64;    // channels per head
constexpr int kG   = 16;    // groups
constexpr int kCPG = 16;    // channels per group
constexpr int kLDA = 68;    // padded LDS row stride (floats); 272B = 17*16 keeps 16B align

__device__ __forceinline__ v8f v8zero() {
  v8f z = {0.f,0.f,0.f,0.f,0.f,0.f,0.f,0.f};
  return z;
}

__device__ __forceinline__ v8f wmma4(v2f a, v2f b, v8f c) {
  // V_WMMA_F32_16X16X4_F32 : D = A(16x4) * B(4x16) + C(16x16), all fp32
  return __builtin_amdgcn_wmma_f32_16x16x4_f32(
      false, a, false, b, (short)0, c, false, false);
}

// Async 16-byte global -> LDS copy (CDNA5 GLOBAL_LOAD_ASYNC_TO_LDS_B128,
// tracked by ASYNCcnt). Builtin signature (from compiler diagnostic):
//   (v4i32 addrspace(1)*, v4i32 addrspace(3)*, imm int offset, imm int cpol)
__device__ __forceinline__ void async_ld16(float* lds, const float* g) {
#if defined(__gfx1250__) && __has_builtin(__builtin_amdgcn_global_load_async_to_lds_b128)
  __builtin_amdgcn_global_load_async_to_lds_b128(
      (__attribute__((address_space(1))) v4i32*)const_cast<float*>(g),
      (__attribute__((address_space(3))) v4i32*)lds, 0, 0);
#else
  *(float4*)lds = *(const float4*)g;
#endif
}

__device__ __forceinline__ void async_wait() {
#if defined(__gfx1250__)
  asm volatile("s_wait_asynccnt 0" ::: "memory");
#endif
}

// ---------------------------------------------------------------------------
// Kernel 1: GroupNorm statistics. One block per (batch, group).
// Group channels are contiguous: offset = (b*256 + g*16)*T = bg*16*T.
// ---------------------------------------------------------------------------
__global__ void __launch_bounds__(256)
gn_stats_kernel(const float* __restrict__ x, float* __restrict__ stats) {
  const int bg = blockIdx.x;                       // [0, 64)
  const float* p = x + (size_t)bg * kCPG * kT;
  float s = 0.f, ss = 0.f;
  for (int i = threadIdx.x; i < kCPG * kT; i += 256) {
    float v = p[i];
    s += v; ss += v * v;
  }
  __shared__ float rs[256], rq[256];
  rs[threadIdx.x] = s; rq[threadIdx.x] = ss;
  __syncthreads();
  for (int off = 128; off > 0; off >>= 1) {
    if (threadIdx.x < off) {
      rs[threadIdx.x] += rs[threadIdx.x + off];
      rq[threadIdx.x] += rq[threadIdx.x + off];
    }
    __syncthreads();
  }
  if (threadIdx.x == 0) {
    const float inv  = 1.0f / (float)(kCPG * kT);
    const float mean = rs[0] * inv;
    const float var  = rq[0] * inv - mean * mean;
    stats[bg * 2 + 0] = mean;
    stats[bg * 2 + 1] = rsqrtf(var + 1e-5f);
  }
}

// ---------------------------------------------------------------------------
// Kernel 2: fused GroupNorm-apply + QKV GEMM + bias + q/k scaling + layout.
//   Qt [bh][t][64]  (q * scale, t-major  -> A operand of Q^T K)
//   Kb [bh][64][T]  (k * scale, natural  -> B operand of Q^T K)
//   Vt [bh][t][64]  (v,        t-major  -> B operand of P V)
// 128 threads = 4 waves; each wave computes a 32x32 quarter (2x2 WMMA tiles).
// ---------------------------------------------------------------------------
__global__ void __launch_bounds__(128)
qkv_kernel(const float* __restrict__ x,  const float* __restrict__ nw,
           const float* __restrict__ nb, const float* __restrict__ qkvw,
           const float* __restrict__ qkvb, const float* __restrict__ stats,
           float* __restrict__ Qt, float* __restrict__ Kb,
           float* __restrict__ Vt) {
  __shared__ __align__(16) float As[64 * kLDA];
  __shared__ __align__(16) float Bs[64 * kLDA];
  const int t0 = blockIdx.x * 64;
  const int o0 = blockIdx.y * 64;        // output row tile in [0, 768)
  const int b  = blockIdx.z;
  const int tid  = threadIdx.x;
  const int wave = tid >> 5, lane = tid & 31;
  const int m0w = (wave & 1) * 32, n0w = (wave >> 1) * 32;
  const int lr = lane & 15, hi = lane >> 4;

  v8f acc[2][2];
  for (int i = 0; i < 2; ++i)
    for (int j = 0; j < 2; ++j) acc[i][j] = v8zero();

  for (int kc = 0; kc < kC; kc += 64) {
    // A: weight tile, async 16B quads (row-contiguous in global and LDS)
    for (int idx = tid; idx < 64 * 16; idx += 128) {
      int r = idx >> 4, c4 = (idx & 15) * 4;
      async_ld16(&As[r * kLDA + c4], &qkvw[(size_t)(o0 + r) * kC + kc + c4]);
    }
    // B: normalized x tile (GroupNorm fused -> must go through VGPRs)
    for (int idx = tid; idx < 64 * 64; idx += 128) {
      int r = idx >> 6, tt = idx & 63;
      int c = kc + r;
      int bg = b * kG + (c >> 4);
      float mean = stats[bg * 2], rstd = stats[bg * 2 + 1];
      float v = x[((size_t)b * kC + c) * kT + t0 + tt];
      Bs[r * kLDA + tt] = (v - mean) * rstd * nw[c] + nb[c];
    }
    async_wait();
    __syncthreads();
    for (int k4 = 0; k4 < 64; k4 += 4) {
      const int ak = k4 + hi * 2;
      v2f a0, a1, b0, b1;
      a0.x = As[(m0w + lr) * kLDA + ak];      a0.y = As[(m0w + lr) * kLDA + ak + 1];
      a1.x = As[(m0w + 16 + lr) * kLDA + ak]; a1.y = As[(m0w + 16 + lr) * kLDA + ak + 1];
      b0.x = Bs[ak * kLDA + n0w + lr];        b0.y = Bs[(ak + 1) * kLDA + n0w + lr];
      b1.x = Bs[ak * kLDA + n0w + 16 + lr];   b1.y = Bs[(ak + 1) * kLDA + n0w + 16 + lr];
      acc[0][0] = wmma4(a0, b0, acc[0][0]);
      acc[0][1] = wmma4(a0, b1, acc[0][1]);
      acc[1][0] = wmma4(a1, b0, acc[1][0]);
      acc[1][1] = wmma4(a1, b1, acc[1][1]);
    }
    __syncthreads();
  }

  const float scale = 0.35355339059327373f;  // 1 / 64^(1/4)
  for (int i = 0; i < 2; ++i)
    for (int j = 0; j < 2; ++j) {
      const int col = n0w + j * 16 + lr;
      const int t = t0 + col;
#pragma unroll
      for (int v = 0; v < 8; ++v) {
        const int row = m0w + i * 16 + v + hi * 8;
        const int o = o0 + row;
        float val = acc[i][j][v] + qkvb[o];
        const int hd = o / 192, r = o % 192;
        const int bh = b * kNH + hd;
        if (r < 64)
          Qt[((size_t)bh * kT + t) * kCH + r] = val * scale;
        else if (r < 128)
          Kb[((size_t)bh * kCH + (r - 64)) * kT + t] = val * scale;
        else
          Vt[((size_t)bh * kT + t) * kCH + (r - 128)] = val;
      }
    }
}

// ---------------------------------------------------------------------------
// Kernel 3: flash attention. One block per (bh, 64-query tile).
// Online softmax; K and V share one LDS buffer (K dead before V is needed).
// Tiles staged with GLOBAL_LOAD_ASYNC_TO_LDS_B128.
// ---------------------------------------------------------------------------
__global__ void __launch_bounds__(128)
attn_kernel(const float* __restrict__ Qt, const float* __restrict__ Kb,
            const float* __restrict__ Vt, float* __restrict__ Ht) {
  __shared__ __align__(16) float Qs[64 * kLDA];
  __shared__ __align__(16) float KVs[64 * kLDA];
  __shared__ __align__(16) float Ss[64 * kLDA];
  __shared__ float mrow[64], lrow[64], frow[64];

  const int t0 = blockIdx.x * 64;
  const int bh = blockIdx.y;
  const int tid  = threadIdx.x;
  const int wave = tid >> 5, lane = tid & 31;
  const int m0w = (wave & 1) * 32, n0w = (wave >> 1) * 32;
  const int lr = lane & 15, hi = lane >> 4;

  // Q tile: rows=t, cols=ch (async; drained at first in-loop wait)
  for (int idx = tid; idx < 64 * 16; idx += 128) {
    int r = idx >> 4, c4 = (idx & 15) * 4;
    async_ld16(&Qs[r * kLDA + c4], &Qt[((size_t)bh * kT + t0 + r) * kCH + c4]);
  }
  if (tid < 64) { mrow[tid] = -1e30f; lrow[tid] = 0.f; }

  v8f oacc[2][2];
  for (int i = 0; i < 2; ++i)
    for (int j = 0; j < 2; ++j) oacc[i][j] = v8zero();

  for (int s0 = 0; s0 < kT; s0 += 64) {
    __syncthreads();                                   // prev iter done with LDS
    for (int idx = tid; idx < 64 * 16; idx += 128) {   // K tile: rows=ch, cols=s
      int r = idx >> 4, c4 = (idx & 15) * 4;
      async_ld16(&KVs[r * kLDA + c4], &Kb[((size_t)bh * kCH + r) * kT + s0 + c4]);
    }
    async_wait();                                      // covers Q on first pass
    __syncthreads();

    v8f sacc[2][2];
    for (int i = 0; i < 2; ++i)
      for (int j = 0; j < 2; ++j) sacc[i][j] = v8zero();
    for (int k4 = 0; k4 < 64; k4 += 4) {               // S = Q^T(t,ch) * K(ch,s)
      const int ak = k4 + hi * 2;
      v2f a0, a1, b0, b1;
      a0.x = Qs[(m0w + lr) * kLDA + ak];      a0.y = Qs[(m0w + lr) * kLDA + ak + 1];
      a1.x = Qs[(m0w + 16 + lr) * kLDA + ak]; a1.y = Qs[(m0w + 16 + lr) * kLDA + ak + 1];
      b0.x = KVs[ak * kLDA + n0w + lr];       b0.y = KVs[(ak + 1) * kLDA + n0w + lr];
      b1.x = KVs[ak * kLDA + n0w + 16 + lr];  b1.y = KVs[(ak + 1) * kLDA + n0w + 16 + lr];
      sacc[0][0] = wmma4(a0, b0, sacc[0][0]);
      sacc[0][1] = wmma4(a0, b1, sacc[0][1]);
      sacc[1][0] = wmma4(a1, b0, sacc[1][0]);
      sacc[1][1] = wmma4(a1, b1, sacc[1][1]);
    }
    for (int i = 0; i < 2; ++i)                        // spill S tile to LDS
      for (int j = 0; j < 2; ++j) {
        const int col = n0w + j * 16 + lr;
#pragma unroll
        for (int v = 0; v < 8; ++v)
          Ss[(m0w + i * 16 + v + hi * 8) * kLDA + col] = sacc[i][j][v];
      }
    __syncthreads();

    // V tile (rows=s, cols=ch) streams in asynchronously while softmax runs
    for (int idx = tid; idx < 64 * 16; idx += 128) {
      int r = idx >> 4, c4 = (idx & 15) * 4;
      async_ld16(&KVs[r * kLDA + c4], &Vt[((size_t)bh * kT + s0 + r) * kCH + c4]);
    }
    if (tid < 64) {                                    // online softmax, one row each
      const int r = tid;
      float mt = -1e30f;
      for (int s = 0; s < 64; ++s) mt = fmaxf(mt, Ss[r * kLDA + s]);
      const float mp = mrow[r];
      const float mn = fmaxf(mp, mt);
      const float f  = __expf(mp - mn);
      float rsum = 0.f;
      for (int s = 0; s < 64; ++s) {
        float p = __expf(Ss[r * kLDA + s] - mn);
        Ss[r * kLDA + s] = p;
        rsum += p;
      }
      lrow[r] = lrow[r] * f + rsum;
      mrow[r] = mn;
      frow[r] = f;
    }
    async_wait();
    __syncthreads();

    for (int i = 0; i < 2; ++i)                        // rescale O accumulators
      for (int j = 0; j < 2; ++j) {
#pragma unroll
        for (int v = 0; v < 8; ++v)
          oacc[i][j][v] *= frow[m0w + i * 16 + v + hi * 8];
      }
    for (int k4 = 0; k4 < 64; k4 += 4) {               // O += P(t,s) * V(s,ch)
      const int ak = k4 + hi * 2;
      v2f a0, a1, b0, b1;
      a0.x = Ss[(m0w + lr) * kLDA + ak];      a0.y = Ss[(m0w + lr) * kLDA + ak + 1];
      a1.x = Ss[(m0w + 16 + lr) * kLDA + ak]; a1.y = Ss[(m0w + 16 + lr) * kLDA + ak + 1];
      b0.x = KVs[ak * kLDA + n0w + lr];       b0.y = KVs[(ak + 1) * kLDA + n0w + lr];
      b1.x = KVs[ak * kLDA + n0w + 16 + lr];  b1.y = KVs[(ak + 1) * kLDA + n0w + 16 + lr];
      oacc[0][0] = wmma4(a0, b0, oacc[0][0]);
      oacc[0][1] = wmma4(a0, b1, oacc[0][1]);
      oacc[1][0] = wmma4(a1, b0, oacc[1][0]);
      oacc[1][1] = wmma4(a1, b1, oacc[1][1]);
    }
  }

  for (int i = 0; i < 2; ++i)                          // normalize + store h^T
    for (int j = 0; j < 2; ++j) {
      const int col = n0w + j * 16 + lr;
#pragma unroll
      for (int v = 0; v < 8; ++v) {
        const int row = m0w + i * 16 + v + hi * 8;
        const float linv = 1.0f / lrow[row];
        Ht[((size_t)bh * kT + t0 + row) * kCH + col] = oacc[i][j][v] * linv;
      }
    }
}

// ---------------------------------------------------------------------------
// Kernel 4: proj GEMM + bias + residual.  h channel c lives at
// Ht[(b*4 + c/64)][t][c%64]; K-chunks are 64-aligned so the head is fixed per
// chunk.  B tile is stored t-major (Bs[n][k]) so global reads are contiguous
// 16B quads (async-able); the B fragment then reads 2 consecutive floats per
// lane, same stride-68 conflict-free pattern as the A fragment.
// ---------------------------------------------------------------------------
__global__ void __launch_bounds__(128)
proj_kernel(const float* __restrict__ x, const float* __restrict__ pw,
            const float* __restrict__ pb, const float* __restrict__ Ht,
            float* __restrict__ out) {
  __shared__ __align__(16) float As[64 * kLDA];
  __shared__ __align__(16) float Bs[64 * kLDA];   // Bs[n][k] (t-major)
  const int t0 = blockIdx.x * 64;
  const int o0 = blockIdx.y * 64;
  const int b  = blockIdx.z;
  const int tid  = threadIdx.x;
  const int wave = tid >> 5, lane = tid & 31;
  const int m0w = (wave & 1) * 32, n0w = (wave >> 1) * 32;
  const int lr = lane & 15, hi = lane >> 4;

  v8f acc[2][2];
  for (int i = 0; i < 2; ++i)
    for (int j = 0; j < 2; ++j) acc[i][j] = v8zero();

  for (int kc = 0; kc < kC; kc += 64) {
    for (int idx = tid; idx < 64 * 16; idx += 128) {   // A: proj_w tile
      int r = idx >> 4, c4 = (idx & 15) * 4;
      async_ld16(&As[r * kLDA + c4], &pw[(size_t)(o0 + r) * kC + kc + c4]);
    }
    const int bh = b * kNH + (kc >> 6);
    for (int idx = tid; idx < 64 * 16; idx += 128) {   // B: h tile, t-major rows
      int n = idx >> 4, c4 = (idx & 15) * 4;
      async_ld16(&Bs[n * kLDA + c4], &Ht[((size_t)bh * kT + t0 + n) * kCH + c4]);
    }
    async_wait();
    __syncthreads();
    for (int k4 = 0; k4 < 64; k4 += 4) {
      const int ak = k4 + hi * 2;
      v2f a0, a1, b0, b1;
      a0.x = As[(m0w + lr) * kLDA + ak];      a0.y = As[(m0w + lr) * kLDA + ak + 1];
      a1.x = As[(m0w + 16 + lr) * kLDA + ak]; a1.y = As[(m0w + 16 + lr) * kLDA + ak + 1];
      b0.x = Bs[(n0w + lr) * kLDA + ak];      b0.y = Bs[(n0w + lr) * kLDA + ak + 1];
      b1.x = Bs[(n0w + 16 + lr) * kLDA + ak]; b1.y = Bs[(n0w + 16 + lr) * kLDA + ak + 1];
      acc[0][0] = wmma4(a0, b0, acc[0][0]);
      acc[0][1] = wmma4(a0, b1, acc[0][1]);
      acc[1][0] = wmma4(a1, b0, acc[1][0]);
      acc[1][1] = wmma4(a1, b1, acc[1][1]);
    }
    __syncthreads();
  }

  for (int i = 0; i < 2; ++i)
    for (int j = 0; j < 2; ++j) {
      const int col = n0w + j * 16 + lr;
      const int t = t0 + col;
#pragma unroll
      for (int v = 0; v < 8; ++v) {
        const int row = m0w + i * 16 + v + hi * 8;
        const int o = o0 + row;
        const size_t g = ((size_t)b * kC + o) * kT + t;
        out[g] = x[g] + acc[i][j][v] + pb[o];
      }
    }
}

// ---------------------------------------------------------------------------
extern "C" void kernel_launch(void* const* d_in, const int* in_sizes, int n_in,
                              void* d_out, int out_size, void* d_ws,
                              size_t ws_size, hipStream_t stream) {
  const float* x    = (const float*)d_in[0];
  const float* nw   = (const float*)d_in[1];
  const float* nb   = (const float*)d_in[2];
  const float* qkvw = (const float*)d_in[3];
  const float* qkvb = (const float*)d_in[4];
  const float* pw   = (const float*)d_in[5];
  const float* pb   = (const float*)d_in[6];
  float* out = (float*)d_out;

  float* ws    = (float*)d_ws;
  float* stats = ws;                                  // 128 floats
  float* Qt    = ws + 128;                            // [16][4096][64]
  float* Kb    = Qt + (size_t)kBH * kT * kCH;         // [16][64][4096]
  float* Vt    = Kb + (size_t)kBH * kCH * kT;         // [16][4096][64]
  float* Ht    = Vt + (size_t)kBH * kT * kCH;         // [16][4096][64]

  gn_stats_kernel<<<dim3(kB * kG), 256, 0, stream>>>(x, stats);
  qkv_kernel<<<dim3(kT / 64, (3 * kC) / 64, kB), 128, 0, stream>>>(
      x, nw, nb, qkvw, qkvb, stats, Qt, Kb, Vt);
  attn_kernel<<<dim3(kT / 64, kBH), 128, 0, stream>>>(Qt, Kb, Vt, Ht);
  proj_kernel<<<dim3(kT / 64, kC / 64, kB), 128, 0, stream>>>(x, pw, pb, Ht, out);
}